// RGATLayer_82729660056049
// MI455X (gfx1250) — compile-verified
//
#include <hip/hip_runtime.h>
#include <hip/hip_bf16.h>

// ---------------------------------------------------------------------------
// RGAT layer for MI455X (gfx1250, wave32).
// GEMMs: V_WMMA_F32_16X16X4_F32 with async-to-LDS double-buffered B panels.
// Shapes: B=64, N=256, D=128, E=8192, R=8.
// ---------------------------------------------------------------------------

typedef float v2f __attribute__((ext_vector_type(2)));
typedef float v8f __attribute__((ext_vector_type(8)));

#define Bc 64
#define Nc 256
#define Dc 128
#define Ec 8192
#define Rc 8

#define KC   32          // K-chunk staged in LDS
#define LDSW 136         // padded row stride (floats): lane-groups hit disjoint banks

// ---------------- ordered-float encoding for atomicMax on uint ----------------
__device__ __forceinline__ unsigned fenc(float f) {
    unsigned u = __float_as_uint(f);
    return (u & 0x80000000u) ? ~u : (u | 0x80000000u);
}
__device__ __forceinline__ float fdec(unsigned u) {
    return __uint_as_float((u & 0x80000000u) ? (u & 0x7fffffffu) : ~u);
}

__device__ __forceinline__ float lrelu(float x) { return x > 0.0f ? x : 0.01f * x; }

// ---------------- zero scratch ----------------
__global__ void k_zero(unsigned* __restrict__ p, long n) {
    long i = (long)blockIdx.x * blockDim.x + threadIdx.x;
    long stride = (long)gridDim.x * blockDim.x;
    for (; i < n; i += stride) p[i] = 0u;
}

// ---------------- per-row dot products (one wave per row, float4 per lane)
__global__ void k_rowdots(const float* __restrict__ h,
                          const float* __restrict__ w_sub_attn,
                          const float* __restrict__ w_attn,
                          float* __restrict__ s_j, float* __restrict__ s_i,
                          float* __restrict__ t_src, float* __restrict__ t_dst,
                          int rows) {
    int row  = blockIdx.x * (blockDim.x >> 5) + (threadIdx.x >> 5);
    int lane = threadIdx.x & 31;
    if (row >= rows) return;
    const float4 hv = ((const float4*)(h + (long)row * Dc))[lane];
    const float4 a0 = ((const float4*)(w_sub_attn))[lane];        // a_nbr
    const float4 a1 = ((const float4*)(w_sub_attn + Dc))[lane];   // a_ctr
    const float4 w0 = ((const float4*)(w_attn))[lane];
    const float4 w1 = ((const float4*)(w_attn + Dc))[lane];
    float p0 = hv.x*a0.x + hv.y*a0.y + hv.z*a0.z + hv.w*a0.w;
    float p1 = hv.x*a1.x + hv.y*a1.y + hv.z*a1.z + hv.w*a1.w;
    float p2 = hv.x*w0.x + hv.y*w0.y + hv.z*w0.z + hv.w*w0.w;
    float p3 = hv.x*w1.x + hv.y*w1.y + hv.z*w1.z + hv.w*w1.w;
    for (int off = 16; off; off >>= 1) {
        p0 += __shfl_xor(p0, off);
        p1 += __shfl_xor(p1, off);
        p2 += __shfl_xor(p2, off);
        p3 += __shfl_xor(p3, off);
    }
    if (lane == 0) { s_j[row] = p0; s_i[row] = p1; t_src[row] = p2; t_dst[row] = p3; }
}

// ---------------- subgraph attention rows (one wave per (b,i) row)
__global__ void k_alpha_sub(const float* __restrict__ s_i, const float* __restrict__ s_j,
                            const int* __restrict__ sg, float* __restrict__ A_sub) {
    int row  = blockIdx.x * (blockDim.x >> 5) + (threadIdx.x >> 5); // b*N+i
    int lane = threadIdx.x & 31;
    if (row >= Bc * Nc) return;
    int b = row / Nc;
    float si = s_i[row];
    int target = (sg[row] == 1) ? 0 : 1;
    const float* sjb = s_j + (long)b * Nc;
    const int*   sgb = sg  + (long)b * Nc;
    float sc[8];
    float mx = -3.0e38f;
    #pragma unroll
    for (int t = 0; t < 8; ++t) {
        int j = lane + t * 32;
        float v = lrelu(si + sjb[j]);
        if (sgb[j] != target) v = -1.0e9f;
        sc[t] = v;
        mx = fmaxf(mx, v);
    }
    for (int off = 16; off; off >>= 1) mx = fmaxf(mx, __shfl_xor(mx, off));
    float sum = 0.0f;
    #pragma unroll
    for (int t = 0; t < 8; ++t) { sc[t] = expf(sc[t] - mx); sum += sc[t]; }
    for (int off = 16; off; off >>= 1) sum += __shfl_xor(sum, off);
    float inv = 1.0f / sum;
    float* Ar = A_sub + (long)row * Nc;
    #pragma unroll
    for (int t = 0; t < 8; ++t) Ar[lane + t * 32] = sc[t] * inv;
}

// ---------------- gate[b,n] = sigmoid( h.wg0 + sub_msg.wg1 )
__global__ void k_gate(const float* __restrict__ h, const float* __restrict__ sub_msg,
                       const float* __restrict__ w_sub_gate, float* __restrict__ gate,
                       int rows) {
    int row  = blockIdx.x * (blockDim.x >> 5) + (threadIdx.x >> 5);
    int lane = threadIdx.x & 31;
    if (row >= rows) return;
    const float4 hv = ((const float4*)(h       + (long)row * Dc))[lane];
    const float4 sv = ((const float4*)(sub_msg + (long)row * Dc))[lane];
    const float4 g0 = ((const float4*)(w_sub_gate))[lane];
    const float4 g1 = ((const float4*)(w_sub_gate + Dc))[lane];
    float p = hv.x*g0.x + hv.y*g0.y + hv.z*g0.z + hv.w*g0.w
            + sv.x*g1.x + sv.y*g1.y + sv.z*g1.z + sv.w*g1.w;
    for (int off = 16; off; off >>= 1) p += __shfl_xor(p, off);
    if (lane == 0) gate[row] = 1.0f / (1.0f + expf(-p));
}

// ---------------- edge logits + segment max (ordered-uint atomicMax)
__global__ void k_edge_e(const float* __restrict__ t_src, const float* __restrict__ t_dst,
                         const int* __restrict__ src, const int* __restrict__ dst,
                         float* __restrict__ e_raw, unsigned* __restrict__ mseg) {
    int idx = blockIdx.x * blockDim.x + threadIdx.x;
    if (idx >= Bc * Ec) return;
    int b = idx / Ec;
    int s = src[idx];
    int d = dst[idx];
    float v = lrelu(t_src[b * Nc + s] + t_dst[b * Nc + d]);
    e_raw[idx] = v;
    atomicMax(&mseg[b * Nc + d], fenc(v));
}

// ---------------- ex = exp(e - m[seg]);  denom[seg] += ex
__global__ void k_edge_ex(const float* __restrict__ e_raw, const unsigned* __restrict__ mseg,
                          const int* __restrict__ dst, float* __restrict__ exbuf,
                          float* __restrict__ denom) {
    int idx = blockIdx.x * blockDim.x + threadIdx.x;
    if (idx >= Bc * Ec) return;
    int b = idx / Ec;
    int seg = b * Nc + dst[idx];
    float ex = expf(e_raw[idx] - fdec(mseg[seg]));
    exbuf[idx] = ex;
    atomicAdd(&denom[seg], ex);
}

// ---------------- T[b,dst,rel,:] += (ex/denom)*norm * h[b,src,:]  (block per edge)
__global__ void k_scatter(const float* __restrict__ h, const float* __restrict__ exbuf,
                          const float* __restrict__ denom, const float* __restrict__ norm,
                          const int* __restrict__ src, const int* __restrict__ dst,
                          const int* __restrict__ rel, float* __restrict__ T) {
    int eidx = blockIdx.x;           // uniform per block -> scalar loads
    int d    = threadIdx.x;          // 0 .. 127
    int b = eidx / Ec;
    int s = src[eidx];
    int dn = dst[eidx];
    int r = rel[eidx];
    int seg = b * Nc + dn;
    float coeff = (exbuf[eidx] / denom[seg]) * norm[eidx];
    long tIdx = (((long)seg) * Rc + r) * Dc + d;
    atomicAdd(&T[tIdx], coeff * h[((long)b * Nc + s) * Dc + d]);
}

// ---------------------------------------------------------------------------
// Batched f32 WMMA GEMM, N fixed at 128:  C = A(MxK) @ B(Kx128) [+bias][*rowscale][+=C]
// Workgroup = 256 threads (8 waves) -> 128x128 output block; each wave owns a
// 16x128 slab (8 v8f accumulators; A fragment reused 8x). B panel is staged in
// LDS in KC=32 chunks via GLOBAL_LOAD_ASYNC_TO_LDS_B128, double buffered and
// paced with s_wait_asynccnt.
// ---------------------------------------------------------------------------
__global__ __launch_bounds__(256)
void wmma_gemm_f32(const float* __restrict__ A, const float* __restrict__ Bm,
                   float* __restrict__ C,
                   int M, int K,
                   long sA, long sB, long sC,
                   const float* __restrict__ bias,
                   const float* __restrict__ rowscale,
                   int accumulate) {
    const int tid  = threadIdx.x;
    const int wave = tid >> 5;
    const int lane = tid & 31;
    const int mBase = blockIdx.x * 128 + wave * 16;

    const float* Ab = A + (long)blockIdx.y * sA;
    const float* Bb = Bm + (long)blockIdx.y * sB;
    float*       Cb = C + (long)blockIdx.y * sC;

    __shared__ __align__(16) float lbuf[2][KC * LDSW];

    // cooperative async copy of one KCx128 B chunk into LDS buffer bi
    auto issue_chunk = [&](int kc, int bi) {
        #pragma unroll
        for (int sswp = 0; sswp < 4; ++sswp) {
            int flat = sswp * 1024 + tid * 4;          // 4 floats per thread per sweep
            int r   = flat >> 7;                       // 0..KC-1
            int col = flat & 127;                      // multiple of 4
            unsigned loff = (unsigned)(unsigned long long)&lbuf[bi][r * LDSW + col];
            const float* g = Bb + (long)(kc + r) * 128 + col;
            asm volatile("global_load_async_to_lds_b128 %0, %1, off"
                         :
                         : "v"(loff), "v"(g)
                         : "memory");
        }
    };

    // ISA 7.12.2 (wave32): A 16x4 f32: lane&15 = M, VGPR pair holds K = kp..kp+1,
    // kp = (lane>>4)*2.  B 4x16 f32: lane&15 = N, same K pairing.
    const int kp = (lane >> 4) << 1;
    const float* ap = Ab + (long)(mBase + (lane & 15)) * K + kp;

    v8f acc[8] = {};
    const int nch = K / KC;

    issue_chunk(0, 0);
    for (int c = 0; c < nch; ++c) {
        if (c + 1 < nch) {
            issue_chunk((c + 1) * KC, (c + 1) & 1);
            asm volatile("s_wait_asynccnt 0x4" ::: "memory"); // chunk c landed
        } else {
            asm volatile("s_wait_asynccnt 0x0" ::: "memory");
        }
        __syncthreads();

        const float* lb = &lbuf[c & 1][0];
        #pragma unroll
        for (int k0 = 0; k0 < KC; k0 += 4) {
            v2f af = *(const v2f*)ap;
            ap += 4;
            const float* lrow = lb + (k0 + kp) * LDSW + (lane & 15);
            #pragma unroll
            for (int nt = 0; nt < 8; ++nt) {
                v2f bf;
                bf.x = lrow[nt * 16];
                bf.y = lrow[nt * 16 + LDSW];
                acc[nt] = __builtin_amdgcn_wmma_f32_16x16x4_f32(
                    /*neg_a=*/false, af, /*neg_b=*/false, bf,
                    /*c_mod=*/(short)0, acc[nt],
                    /*reuse_a=*/false, /*reuse_b=*/false);
            }
        }
        __syncthreads();   // safe to refill this buffer two chunks later
    }

    // C/D 16x16 f32: row = g + 8*(lane>>4), col = lane&15
    const int rb = (lane >> 4) << 3;
    #pragma unroll
    for (int nt = 0; nt < 8; ++nt) {
        int col = nt * 16 + (lane & 15);
        float bv = bias ? bias[col] : 0.0f;
        #pragma unroll
        for (int g = 0; g < 8; ++g) {
            int row = mBase + rb + g;
            float v = acc[nt][g] + bv;
            if (rowscale)   v *= rowscale[row];
            long ci = (long)row * 128 + col;
            if (accumulate) v += Cb[ci];
            Cb[ci] = v;
        }
    }
}

// ---------------------------------------------------------------------------
extern "C" void kernel_launch(void* const* d_in, const int* in_sizes, int n_in,
                              void* d_out, int out_size, void* d_ws, size_t ws_size,
                              hipStream_t stream) {
    const float* h          = (const float*)d_in[0];   // (B,N,D)
    const float* norm       = (const float*)d_in[1];   // (B,E)
    const float* weight     = (const float*)d_in[2];   // (R,D,D) -> viewed (R*D, D)
    const float* w_sub_attn = (const float*)d_in[3];   // (2D,1)
    const float* w_sub_gate = (const float*)d_in[4];   // (2D,1)
    const float* W_sub_proj = (const float*)d_in[5];   // (D,D)
    const float* b_sub_proj = (const float*)d_in[6];   // (D,)
    const float* w_attn     = (const float*)d_in[7];   // (2D,1)
    const float* W_self     = (const float*)d_in[8];   // (D,D)
    const float* b_self     = (const float*)d_in[9];   // (D,)
    const int*   sg         = (const int*)d_in[10];    // (B,N)
    const int*   src        = (const int*)d_in[11];    // (B,E)
    const int*   dst        = (const int*)d_in[12];    // (B,E)
    const int*   rel        = (const int*)d_in[13];    // (B,E)
    float*       out        = (float*)d_out;           // (B,N,D)

    // ---- workspace layout (floats) ----
    float* ws = (float*)d_ws;
    long off = 0;
    float* s_j     = ws + off; off += (long)Bc * Nc;
    float* s_i     = ws + off; off += (long)Bc * Nc;
    float* t_src   = ws + off; off += (long)Bc * Nc;
    float* t_dst   = ws + off; off += (long)Bc * Nc;
    float* gate    = ws + off; off += (long)Bc * Nc;
    float* e_raw   = ws + off; off += (long)Bc * Ec;
    float* exbuf   = ws + off; off += (long)Bc * Ec;
    float* A_sub   = ws + off; off += (long)Bc * Nc * Nc;
    float* sub_msg = ws + off; off += (long)Bc * Nc * Dc;
    // contiguous zero-init region: mseg | denom | T
    unsigned* mseg = (unsigned*)(ws + off); off += (long)Bc * Nc;
    float* denom   = ws + off; off += (long)Bc * Nc;
    float* T       = ws + off; off += (long)Bc * Nc * Rc * Dc;
    long zeroWords = (long)Bc * Nc * 2 + (long)Bc * Nc * Rc * Dc;

    const int rows  = Bc * Nc;      // 16384
    const int edges = Bc * Ec;      // 524288

    // 1) zero mseg/denom/T
    k_zero<<<4096, 256, 0, stream>>>(mseg, zeroWords);

    // 2) four per-row dot products
    k_rowdots<<<rows / 8, 256, 0, stream>>>(h, w_sub_attn, w_attn, s_j, s_i, t_src, t_dst, rows);

    // 3) subgraph softmax rows
    k_alpha_sub<<<rows / 8, 256, 0, stream>>>(s_i, s_j, sg, A_sub);

    // 4) sub_msg = A_sub @ h  (batched: M=256, K=256, N=128)
    {
        dim3 grid(Nc / 128, Bc, 1);
        wmma_gemm_f32<<<grid, 256, 0, stream>>>(A_sub, h, sub_msg,
                                                Nc, Nc,
                                                (long)Nc * Nc, (long)Nc * Dc, (long)Nc * Dc,
                                                nullptr, nullptr, 0);
    }

    // 5) gate
    k_gate<<<rows / 8, 256, 0, stream>>>(h, sub_msg, w_sub_gate, gate, rows);

    // 6) edge logits + segment max
    k_edge_e<<<edges / 256, 256, 0, stream>>>(t_src, t_dst, src, dst, e_raw, mseg);

    // 7) exp + segment sum
    k_edge_ex<<<edges / 256, 256, 0, stream>>>(e_raw, mseg, dst, exbuf, denom);

    // 8) weighted scatter into T[b,dst,rel,:]
    k_scatter<<<edges, Dc, 0, stream>>>(h, exbuf, denom, norm, src, dst, rel, T);

    // 9) out = T (16384 x 1024) @ weight_flat (1024 x 128)   (overwrite)
    wmma_gemm_f32<<<dim3(rows / 128, 1, 1), 256, 0, stream>>>(T, weight, out,
                                                              rows, Rc * Dc,
                                                              0, 0, 0,
                                                              nullptr, nullptr, 0);

    // 10) out += h @ W_self + b_self
    wmma_gemm_f32<<<dim3(rows / 128, 1, 1), 256, 0, stream>>>(h, W_self, out,
                                                              rows, Dc,
                                                              0, 0, 0,
                                                              b_self, nullptr, 1);

    // 11) out += (sub_msg @ W_sub_proj + b_sub_proj) * gate
    wmma_gemm_f32<<<dim3(rows / 128, 1, 1), 256, 0, stream>>>(sub_msg, W_sub_proj, out,
                                                              rows, Dc,
                                                              0, 0, 0,
                                                              b_sub_proj, gate, 1);

    (void)in_sizes; (void)n_in; (void)out_size; (void)ws_size;
}